// PGDNetBaseline_62277025792090
// MI455X (gfx1250) — compile-verified
//
#include <hip/hip_runtime.h>
#include <stdint.h>

// PGD beamforming unroll: B=131072 independent problems, M=8, K=4, 10 layers.
// Roofline: ~3 GFLOP total vs ~100MB traffic (4.3us @23.3TB/s) => VALU-bound.
// One element per lane, all state in VGPRs across the 10-layer loop.
// CDNA5 paths: global_load_async_to_lds_b128 (+s_wait_asynccnt) staging, and
// V_PK_FMA_F32 packed math, vectorized over j-columns (two j-pairs).
// Minus signs handled with dual accumulators (one pk_sub at the end) so no
// negated-operand arrays get hoisted -> stays under 256 VGPRs (no MSB mode).
// WMMA deliberately NOT used: 4x4x8 batch-unique matmuls map to 16x16 WMMA only
// via block-diagonal packing (<=25% util) + per-layer cross-lane transposes,
// which costs more than the FMAs it saves.

#define NLAYERS   10
#define NOISE_VAR 0.1f
#define INV_LN2   1.4426950408889634f

#define ELEM_BYTES   128            // 32 floats per array per element
#define SLOT_BYTES   144            // padded LDS slot (36 DWORDs -> conflict-free b128)
#define WAVES_PER_BLK 2
#define BLK (WAVES_PER_BLK * 32)
#define SMEM_BYTES (WAVES_PER_BLK * 4 * 32 * SLOT_BYTES)   // 36864 B

typedef float v2f __attribute__((ext_vector_type(2)));

__device__ __forceinline__ v2f pkfma(v2f a, v2f b, v2f c) {
    return __builtin_elementwise_fma(a, b, c);   // -> v_pk_fma_f32
}
__device__ __forceinline__ v2f splat(float x) { return (v2f){x, x}; }

__device__ __forceinline__ void async_g2l_b128(uint32_t lds_addr, uint64_t gaddr) {
    asm volatile("global_load_async_to_lds_b128 %0, %1, off"
                 :: "v"(lds_addr), "v"(gaddr) : "memory");
}

__global__ __launch_bounds__(BLK, 4) void pgd_kernel(
    const float* __restrict__ g_wr, const float* __restrict__ g_wi,
    const float* __restrict__ g_Hr, const float* __restrict__ g_Hi,
    const float* __restrict__ steps, float* __restrict__ out, int B)
{
    __shared__ __align__(16) unsigned char smem[SMEM_BYTES];

    const int tid  = threadIdx.x;
    const int lane = tid & 31;
    const int wv   = tid >> 5;
    const int waveElem = blockIdx.x * BLK + wv * 32;
    const int elem     = blockIdx.x * BLK + tid;

    const uint32_t smemBase = (uint32_t)(uintptr_t)(&smem[0]);
    const size_t maxByte = (size_t)B * ELEM_BYTES - 16;

    // ---------------- async stage: 4 arrays x (32 elems x 128B) per wave -----
    const float* gsrc[4] = {g_wr, g_wi, g_Hr, g_Hi};
    #pragma unroll
    for (int a = 0; a < 4; ++a) {
        const uint64_t gbase = (uint64_t)(uintptr_t)gsrc[a];
        const uint32_t ldsArr = smemBase + (uint32_t)((wv * 4 + a) * 32) * SLOT_BYTES;
        #pragma unroll
        for (int i = 0; i < 8; ++i) {
            uint32_t p   = (uint32_t)(i * 32 + lane);
            uint32_t e   = p >> 3;
            uint32_t pos = (p & 7u) * 16u;
            size_t   go  = (size_t)waveElem * ELEM_BYTES + (size_t)p * 16u;
            if (go > maxByte) go = maxByte;
            async_g2l_b128(ldsArr + e * SLOT_BYTES + pos, gbase + go);
        }
    }
    asm volatile("s_wait_asynccnt 0x0" ::: "memory");   // own wave's data only

    // ---------------- LDS -> registers -------------------------------------
    // j-pair vectorization: index [m][jp], jp=0 -> j={0,1}, jp=1 -> j={2,3}
    v2f   wr[8][2], wi[8][2];
    float Hr[4][8], Hi[4][8];
    {
        const unsigned char* s0 = smem + ((wv * 4 + 0) * 32 + lane) * SLOT_BYTES;
        const unsigned char* s1 = smem + ((wv * 4 + 1) * 32 + lane) * SLOT_BYTES;
        const unsigned char* s2 = smem + ((wv * 4 + 2) * 32 + lane) * SLOT_BYTES;
        const unsigned char* s3 = smem + ((wv * 4 + 3) * 32 + lane) * SLOT_BYTES;
        #pragma unroll
        for (int q = 0; q < 8; ++q) {                    // w: (M=8,K=4) row-major
            float4 a = *(const float4*)(s0 + q * 16);
            float4 b = *(const float4*)(s1 + q * 16);
            wr[q][0] = (v2f){a.x, a.y}; wr[q][1] = (v2f){a.z, a.w};
            wi[q][0] = (v2f){b.x, b.y}; wi[q][1] = (v2f){b.z, b.w};
        }
        #pragma unroll
        for (int q = 0; q < 8; ++q) {                    // H: (K=4,M=8) row-major
            int k = q >> 1, m0 = (q & 1) * 4;
            float4 a = *(const float4*)(s2 + q * 16);
            Hr[k][m0+0] = a.x; Hr[k][m0+1] = a.y; Hr[k][m0+2] = a.z; Hr[k][m0+3] = a.w;
            float4 b = *(const float4*)(s3 + q * 16);
            Hi[k][m0+0] = b.x; Hi[k][m0+1] = b.y; Hi[k][m0+2] = b.z; Hi[k][m0+3] = b.w;
        }
    }

    // ---------------- 10 PGD layers, all in registers -----------------------
    for (int layer = 0; layer < NLAYERS; ++layer) {
        const float step = steps[layer];                 // uniform -> scalar load

        // hw[k][j] = sum_m H[k][m]*w[m][j] (complex), vectorized over j-pairs.
        // hw_re = (sum Hr*wr) - (sum Hi*wi); hw_im = sum Hr*wi + sum Hi*wr
        v2f hwr[4][2], hwi[4][2];
        #pragma unroll
        for (int k = 0; k < 4; ++k)
            #pragma unroll
            for (int jp = 0; jp < 2; ++jp) {
                v2f a0 = (v2f){0.f, 0.f};   // sum Hr*wr
                v2f a1 = (v2f){0.f, 0.f};   // sum Hi*wi
                v2f ai = (v2f){0.f, 0.f};
                #pragma unroll
                for (int m = 0; m < 8; ++m) {
                    a0 = pkfma(splat(Hr[k][m]), wr[m][jp], a0);
                    a1 = pkfma(splat(Hi[k][m]), wi[m][jp], a1);
                    ai = pkfma(splat(Hr[k][m]), wi[m][jp], ai);
                    ai = pkfma(splat(Hi[k][m]), wr[m][jp], ai);
                }
                hwr[k][jp] = a0 - a1;
                hwi[k][jp] = ai;
            }

        // per-row SINR terms: T_k = noise + sum_j |hw|^2, D_k = T_k - |hw_kk|^2
        float cdiag[4], coff[4];
        #pragma unroll
        for (int k = 0; k < 4; ++k) {
            v2f s = (v2f){NOISE_VAR, 0.f};
            s = pkfma(hwr[k][0], hwr[k][0], pkfma(hwi[k][0], hwi[k][0], s));
            s = pkfma(hwr[k][1], hwr[k][1], pkfma(hwi[k][1], hwi[k][1], s));
            float T  = s.x + s.y;
            float dr = hwr[k][k >> 1][k & 1];
            float di = hwi[k][k >> 1][k & 1];
            float D  = T - fmaf(dr, dr, di * di);
            float invT = 1.0f / T;
            float invD = 1.0f / D;
            cdiag[k] = -2.0f * INV_LN2 * invT;
            coff[k]  = -2.0f * INV_LN2 * (invT - invD);
        }

        // G = c .* hw (in place); c pair = coff everywhere, cdiag at j==k slot
        #pragma unroll
        for (int k = 0; k < 4; ++k) {
            v2f cj0 = splat(coff[k]);
            v2f cj1 = splat(coff[k]);
            if ((k >> 1) == 0) cj0[k & 1] = cdiag[k]; else cj1[k & 1] = cdiag[k];
            hwr[k][0] *= cj0; hwi[k][0] *= cj0;
            hwr[k][1] *= cj1; hwi[k][1] *= cj1;
        }

        // gw[m][j] = sum_k conj(H[k][m])*G[k][j]:
        //   gw_re = sum Hr*Gr + sum Hi*Gi
        //   gw_im = (sum Hr*Gi) - (sum Hi*Gr)
        // then w -= step*gw; accumulate ||w||^2
        v2f nacc = (v2f){0.f, 0.f};
        const v2f negstep = splat(-step);
        #pragma unroll
        for (int m = 0; m < 8; ++m)
            #pragma unroll
            for (int jp = 0; jp < 2; ++jp) {
                v2f gr = (v2f){0.f, 0.f};
                v2f g0 = (v2f){0.f, 0.f};   // sum Hr*Gi
                v2f g1 = (v2f){0.f, 0.f};   // sum Hi*Gr
                #pragma unroll
                for (int k = 0; k < 4; ++k) {
                    gr = pkfma(splat(Hr[k][m]), hwr[k][jp], gr);
                    gr = pkfma(splat(Hi[k][m]), hwi[k][jp], gr);
                    g0 = pkfma(splat(Hr[k][m]), hwi[k][jp], g0);
                    g1 = pkfma(splat(Hi[k][m]), hwr[k][jp], g1);
                }
                v2f gi = g0 - g1;
                wr[m][jp] = pkfma(negstep, gr, wr[m][jp]);
                wi[m][jp] = pkfma(negstep, gi, wi[m][jp]);
                nacc = pkfma(wr[m][jp], wr[m][jp], nacc);
                nacc = pkfma(wi[m][jp], wi[m][jp], nacc);
            }

        // project onto ||W||_F <= sqrt(P_MAX)=1
        float nrm   = sqrtf(nacc.x + nacc.y);
        float scale = fminf(1.0f, 1.0f / fmaxf(nrm, 1e-12f));
        const v2f vs = splat(scale);
        #pragma unroll
        for (int m = 0; m < 8; ++m)
            #pragma unroll
            for (int jp = 0; jp < 2; ++jp) { wr[m][jp] *= vs; wi[m][jp] *= vs; }
    }

    // ---------------- store interleaved complex64 (B,8,4) -------------------
    if (elem < B) {
        float* o = out + (size_t)elem * 64;
        #pragma unroll
        for (int m = 0; m < 8; ++m)
            #pragma unroll
            for (int jp = 0; jp < 2; ++jp) {
                float4 v = make_float4(wr[m][jp].x, wi[m][jp].x,
                                       wr[m][jp].y, wi[m][jp].y);
                *(float4*)(o + m * 8 + jp * 4) = v;
            }
    }
}

extern "C" void kernel_launch(void* const* d_in, const int* in_sizes, int n_in,
                              void* d_out, int out_size, void* d_ws, size_t ws_size,
                              hipStream_t stream) {
    const float* wr    = (const float*)d_in[0];
    const float* wi    = (const float*)d_in[1];
    const float* Hr    = (const float*)d_in[2];
    const float* Hi    = (const float*)d_in[3];
    const float* steps = (const float*)d_in[4];
    const int B = in_sizes[0] / 32;               // (B, 8, 4)
    const int grid = (B + BLK - 1) / BLK;
    pgd_kernel<<<grid, BLK, 0, stream>>>(wr, wi, Hr, Hi, steps, (float*)d_out, B);
}